// PacketTinyModel_82755429859679
// MI455X (gfx1250) — compile-verified
//
#include <hip/hip_runtime.h>

// Tiny linear RNN, B=2048 T=4096 D=8, output f32 [B,T,8].
// Strategy (memory-bound, ~320MB total traffic ~ 13.7us @ 23.3TB/s):
//   init:   fold w_in into embed table (19x8), compute P = (w_state^T)^CHUNK
//   phase1: per (16-batch tile, 64-step chunk) run recurrence from zero state
//           via V_WMMA_F32_16X16X4_F32; store only chunk-end state (8 f32/row)
//   phase2: sequential scan over 64 chunk carries per batch row using P
//   phase3: re-run each chunk from its true carry, write hidden once
// 8192 independent waves in phases 1/3 hide the dependent WMMA chain.

#define VOCAB 19
#define BATCH 2048
#define SEQ 4096
#define CHUNK 64
#define NCHUNK (SEQ / CHUNK)   // 64
#define NTILE (BATCH / 16)     // 128

typedef float v2f __attribute__((ext_vector_type(2)));
typedef float v8f __attribute__((ext_vector_type(8)));

__device__ __forceinline__ float lane_gather(float v, int srcLane) {
  return __builtin_bit_cast(
      float, __builtin_amdgcn_ds_bpermute(srcLane << 2, __builtin_bit_cast(int, v)));
}

// ws layout (floats): [0..151] xin_table (19x8), [160..223] P = (w_state^T)^CHUNK,
// [256 ..) carries[NCHUNK][BATCH][8], then locals[NCHUNK][BATCH][8].
#define WS_TAB 0
#define WS_P 160
#define WS_CARRY 256
#define WS_LOCAL (WS_CARRY + (size_t)NCHUNK * BATCH * 8)

__global__ __launch_bounds__(256) void init_tables(const float* __restrict__ embed,
                                                   const float* __restrict__ w_in,
                                                   const float* __restrict__ w_state,
                                                   float* __restrict__ ws) {
  __shared__ float Wt[64];  // Wt[d][e] = w_state[e][d]  (row-vector step matrix)
  __shared__ float Pw[64];
  const int tid = threadIdx.x;
  if (tid < VOCAB * 8) {  // xin_table[v][e] = sum_d embed[v][d] * w_in[e][d]
    const int v = tid >> 3, e = tid & 7;
    float acc = 0.f;
#pragma unroll
    for (int d = 0; d < 8; ++d) acc += embed[v * 8 + d] * w_in[e * 8 + d];
    ws[WS_TAB + tid] = acc;
  }
  if (tid < 64) {
    const int d = tid >> 3, e = tid & 7;
    const float wt = w_state[e * 8 + d];
    Wt[tid] = wt;
    Pw[tid] = wt;  // power 1
  }
  __syncthreads();
  for (int it = 1; it < CHUNK; ++it) {  // Pw = Wt^CHUNK
    float acc = 0.f;
    if (tid < 64) {
      const int d = tid >> 3, e = tid & 7;
#pragma unroll
      for (int j = 0; j < 8; ++j) acc += Pw[d * 8 + j] * Wt[j * 8 + e];
    }
    __syncthreads();
    if (tid < 64) Pw[tid] = acc;
    __syncthreads();
  }
  if (tid < 64) ws[WS_P + tid] = Pw[tid];
}

// Core chunk recurrence. One wave handles 16 batch rows x CHUNK steps.
// S (8 dims x 16 batches) lives as the WMMA B matrix; A = w_state (rows 8..15 = 0);
// C = gathered input U_t. D = A*B + C computed as two K=4 f32 WMMAs.
//
// Note on lanes 16..31: since A rows 8..15 are zero, D rows 8..15 (which live in
// lanes 16..31 of the C/D VGPRs) equal C rows 8..15. Those lanes' D values are
// never consumed (bpermute sources only lanes 0..15; stores are lane-masked), and
// the lane>=16 table gathers read valid table entries anyway (same n, same LDS
// address -> broadcast). So C needs NO zeroing of its upper rows: ds_load feeds
// the WMMA directly, shortening the per-step dependency chain.
template <bool PHASE3>
__global__ __launch_bounds__(128) void recur_kernel(const int* __restrict__ ids,
                                                    const float* __restrict__ w_state,
                                                    const float* __restrict__ ws,
                                                    float* __restrict__ out) {
  __shared__ alignas(16) float s_tab[VOCAB * 8];
  for (int i = threadIdx.x; i < VOCAB * 8; i += blockDim.x) s_tab[i] = ws[WS_TAB + i];
  __syncthreads();

  const float* __restrict__ carries = ws + WS_CARRY;
  float* __restrict__ locals = const_cast<float*>(ws) + WS_LOCAL;

  const int lane = threadIdx.x & 31;
  const int g = blockIdx.x * 4 + (threadIdx.x >> 5);  // wave linear id, 8192 total
  const int tile = g / NCHUNK;
  const int chunk = g % NCHUNK;
  const int n = lane & 15;  // batch column within tile (lanes 16..31 mirror 0..15)
  const int b = tile * 16 + n;
  const int t0 = chunk * CHUNK;
  const bool lo = lane < 16;
  const int koff = lo ? 0 : 2;  // A/B lane-half K offset (ISA 16x4 f32 layout)

  // A matrix: A[m][k] = w_state[m][k] for m<8, else 0. Lane holds row m=n.
  v2f a_lo = {0.f, 0.f}, a_hi = {0.f, 0.f};
  if (n < 8) {
    a_lo.x = w_state[n * 8 + koff + 0];
    a_lo.y = w_state[n * 8 + koff + 1];
    a_hi.x = w_state[n * 8 + koff + 4];
    a_hi.y = w_state[n * 8 + koff + 5];
  }

  // B matrix init: S[k][n]. Phase1: zeros. Phase3: chunk carry-in state.
  v2f b_lo = {0.f, 0.f}, b_hi = {0.f, 0.f};
  if (PHASE3) {
    const float* cr = carries + ((size_t)chunk * BATCH + b) * 8;
    b_lo.x = cr[koff + 0];
    b_lo.y = cr[koff + 1];
    b_hi.x = cr[koff + 4];
    b_hi.y = cr[koff + 5];
  }

  const int* __restrict__ idrow = ids + (size_t)b * SEQ + t0;
  float* __restrict__ orow = out + ((size_t)b * SEQ + t0) * 8;

  v8f d = {0.f, 0.f, 0.f, 0.f, 0.f, 0.f, 0.f, 0.f};
  for (int s4 = 0; s4 < CHUNK; s4 += 4) {
    const int4 id4 = *(const int4*)(idrow + s4);  // 16B/row per 4 steps; ids fit in L2
    const int idv[4] = {id4.x, id4.y, id4.z, id4.w};
#pragma unroll
    for (int j = 0; j < 4; ++j) {
      const int id = idv[j];
      // C = U_t: C-layout vgpr r, lanes 0..15 -> U[r][n]; upper rows left as-is.
      const float4 u0 = *(const float4*)(s_tab + id * 8);
      const float4 u1 = *(const float4*)(s_tab + id * 8 + 4);
      v8f c;
      c[0] = u0.x;
      c[1] = u0.y;
      c[2] = u0.z;
      c[3] = u0.w;
      c[4] = u1.x;
      c[5] = u1.y;
      c[6] = u1.z;
      c[7] = u1.w;
      // S' = W*S + U, K=8 split into two K=4 WMMAs (EXEC is all-ones here).
      d = __builtin_amdgcn_wmma_f32_16x16x4_f32(false, a_lo, false, b_lo, (short)0, c,
                                                false, false);
      d = __builtin_amdgcn_wmma_f32_16x16x4_f32(false, a_hi, false, b_hi, (short)0, d,
                                                false, false);
      if (PHASE3) {
        if (lo) {  // lane n holds the full 8-dim state of batch b at this t
          const float4 o0 = {d[0], d[1], d[2], d[3]};
          const float4 o1 = {d[4], d[5], d[6], d[7]};
          *(float4*)(orow + (size_t)(s4 + j) * 8) = o0;
          *(float4*)(orow + (size_t)(s4 + j) * 8 + 4) = o1;
        }
      }
      // D (C-layout) -> next-step B: rows {2,3,6,7} must move to lanes 16..31.
      const float s2 = lane_gather(d[2], n);
      const float s3 = lane_gather(d[3], n);
      const float s6 = lane_gather(d[6], n);
      const float s7 = lane_gather(d[7], n);
      b_lo.x = lo ? d[0] : s2;
      b_lo.y = lo ? d[1] : s3;
      b_hi.x = lo ? d[4] : s6;
      b_hi.y = lo ? d[5] : s7;
    }
  }
  if (!PHASE3) {  // store chunk-end local state (zero-start) for the carry scan
    if (lo) {
      float* dst = locals + ((size_t)chunk * BATCH + b) * 8;
      const float4 o0 = {d[0], d[1], d[2], d[3]};
      const float4 o1 = {d[4], d[5], d[6], d[7]};
      *(float4*)dst = o0;
      *(float4*)(dst + 4) = o1;
    }
  }
}

// Sequential scan over chunk carries: carry[k+1] = carry[k] @ P + local[k].
__global__ __launch_bounds__(256) void carry_scan(float* __restrict__ ws) {
  __shared__ float P[64];
  if (threadIdx.x < 64) P[threadIdx.x] = ws[WS_P + threadIdx.x];
  __syncthreads();
  const int b = blockIdx.x * blockDim.x + threadIdx.x;
  if (b >= BATCH) return;
  float* __restrict__ carries = ws + WS_CARRY;
  const float* __restrict__ locals = ws + WS_LOCAL;
  float c[8];
#pragma unroll
  for (int e = 0; e < 8; ++e) c[e] = 0.f;
  for (int k = 0; k < NCHUNK; ++k) {
    float* dst = carries + ((size_t)k * BATCH + b) * 8;
#pragma unroll
    for (int e = 0; e < 8; ++e) dst[e] = c[e];
    const float* l = locals + ((size_t)k * BATCH + b) * 8;
    float nc[8];
#pragma unroll
    for (int e = 0; e < 8; ++e) {
      float acc = l[e];
#pragma unroll
      for (int dd = 0; dd < 8; ++dd) acc += c[dd] * P[dd * 8 + e];
      nc[e] = acc;
    }
#pragma unroll
    for (int e = 0; e < 8; ++e) c[e] = nc[e];
  }
}

extern "C" void kernel_launch(void* const* d_in, const int* in_sizes, int n_in,
                              void* d_out, int out_size, void* d_ws, size_t ws_size,
                              hipStream_t stream) {
  const int* ids = (const int*)d_in[0];
  const float* embed = (const float*)d_in[1];
  const float* w_in = (const float*)d_in[2];
  const float* w_state = (const float*)d_in[3];
  float* out = (float*)d_out;
  float* ws = (float*)d_ws;  // needs WS_LOCAL + NCHUNK*BATCH*8 floats (~8.4 MB)

  init_tables<<<1, 256, 0, stream>>>(embed, w_in, w_state, ws);
  recur_kernel<false><<<NTILE * NCHUNK / 4, 128, 0, stream>>>(ids, w_state, ws, out);
  carry_scan<<<BATCH / 256, 256, 0, stream>>>(ws);
  recur_kernel<true><<<NTILE * NCHUNK / 4, 128, 0, stream>>>(ids, w_state, ws, out);
}